// TextureBaker_84662395339141
// MI455X (gfx1250) — compile-verified
//
#include <hip/hip_runtime.h>
#include <cmath>

typedef __attribute__((ext_vector_type(2))) float v2f;
typedef __attribute__((ext_vector_type(8))) float v8f;

#define BAKE_EPS 1e-12f
#define FID_NONE 0x7FFFFFFF

// ---------------------------------------------------------------------------
// Kernel 1: per-triangle affine barycentric coefficients.
// For triangle t, coord q in {u,v,w}: coef[(t*3+q)*4 + {0,1,2,3}] = {Gx,Gy,h,0}
// so that bary_q(p) = Gx*px + Gy*py + h  (== reference _bary output exactly in
// exact arithmetic). Degenerate triangles get G=0, h=-1 (never "inside").
// ---------------------------------------------------------------------------
__global__ void bake_coef_kernel(const float* __restrict__ uv,
                                 const int* __restrict__ faces,
                                 float* __restrict__ coef, int nTri) {
    int t = blockIdx.x * blockDim.x + threadIdx.x;
    if (t >= nTri) return;
    int ia = faces[3 * t + 0], ib = faces[3 * t + 1], ic = faces[3 * t + 2];
    float ax = uv[2 * ia], ay = uv[2 * ia + 1];
    float bx = uv[2 * ib], by = uv[2 * ib + 1];
    float cx = uv[2 * ic], cy = uv[2 * ic + 1];
    float v0x = bx - ax, v0y = by - ay;
    float v1x = cx - ax, v1y = cy - ay;
    float d00 = v0x * v0x + v0y * v0y;
    float d01 = v0x * v1x + v0y * v1y;
    float d11 = v1x * v1x + v1y * v1y;
    float denom = d00 * d11 - d01 * d01;
    bool bad = fabsf(denom) < BAKE_EPS;
    float ds = bad ? 1.0f : denom;
    float gvx = (d11 * v0x - d01 * v1x) / ds, gvy = (d11 * v0y - d01 * v1y) / ds;
    float gwx = (d00 * v1x - d01 * v0x) / ds, gwy = (d00 * v1y - d01 * v0y) / ds;
    float hv = -(gvx * ax + gvy * ay);
    float hw = -(gwx * ax + gwy * ay);
    float gux = -(gvx + gwx), guy = -(gvy + gwy);
    float hu = 1.0f - hv - hw;
    if (bad) { gux = guy = gvx = gvy = gwx = gwy = 0.0f; hu = hv = hw = -1.0f; }
    float* c = coef + (size_t)t * 12;
    c[0] = gux; c[1] = guy; c[2]  = hu; c[3]  = 0.0f;
    c[4] = gvx; c[5] = gvy; c[6]  = hv; c[7]  = 0.0f;
    c[8] = gwx; c[9] = gwy; c[10] = hw; c[11] = 0.0f;
}

// ---------------------------------------------------------------------------
// Kernel 2: one wave32 handles 16 pixels.
//   A (16x4, 2 VGPRs): row m = [x_m, y_m, 1, 0]
//     lanes 0-15 : VGPR0 = K0 (x), VGPR1 = K1 (y)   (m = lane)
//     lanes 16-31: VGPR0 = K2 (1), VGPR1 = K3 (0)   (m = lane-16)
//   B (4x16, 2 VGPRs): col n = [Gx, Gy, h, 0] of triangle chunk*16+n
//     lanes load coef[... + 2*g], coef[... + 2*g + 1]  (g = lane>>4)
//   D (16x16 f32): lane l, VGPR r  ->  pixel m = r + 8*(l>>4), triangle n = l&15
// inside = (u>=0 && v>=0 && w>=0) == (min3(u,v,w) >= 0)  -> v_min3_num_f32.
// First hit == min hit index -> running min over chunks, then lane-min within
// 16-lane groups, LDS handoff, and exact reference interpolation on lanes 0-15.
// ---------------------------------------------------------------------------
__global__ void __launch_bounds__(256)
bake_raster_kernel(const float* __restrict__ attr,
                   const float* __restrict__ uv,
                   const int* __restrict__ faces,
                   const float* __restrict__ coef,
                   float* __restrict__ out, int res, int nTri) {
    const int lane = threadIdx.x & 31;
    const int waveInBlk = threadIdx.x >> 5;
    const int waveId = blockIdx.x * (blockDim.x >> 5) + waveInBlk;
    const int total = res * res;
    const int pixBase = waveId * 16;

    __shared__ int sfid[8 * 16];

    const int m = lane & 15;      // pixel row of A this lane supplies
    const int g = lane >> 4;      // half-wave group
    int pc = pixBase + m;
    if (pc >= total) pc = total - 1;             // branchless clamp (keeps EXEC full)
    const float invres = 1.0f / (float)res;
    const float px = ((float)(pc % res) + 0.5f) * invres;
    const float py = ((float)(pc / res) + 0.5f) * invres;

    v2f A;
    A.x = g ? 1.0f : px;   // K = 2*g
    A.y = g ? 0.0f : py;   // K = 2*g + 1

    int fid[8];
#pragma unroll
    for (int r = 0; r < 8; ++r) fid[r] = FID_NONE;

    const v8f zero = {};
    const int n = lane & 15;      // triangle column this lane owns in D
    for (int c = 0; c * 16 < nTri; ++c) {
        const int t = c * 16 + n;
        const float* base = coef + (size_t)t * 12 + 2 * g;
        v2f Bu = *(const v2f*)(base + 0);
        v2f Bv = *(const v2f*)(base + 4);
        v2f Bw = *(const v2f*)(base + 8);
        v8f du = __builtin_amdgcn_wmma_f32_16x16x4_f32(false, A, false, Bu,
                                                       (short)0, zero, false, false);
        v8f dv = __builtin_amdgcn_wmma_f32_16x16x4_f32(false, A, false, Bv,
                                                       (short)0, zero, false, false);
        v8f dw = __builtin_amdgcn_wmma_f32_16x16x4_f32(false, A, false, Bw,
                                                       (short)0, zero, false, false);
#pragma unroll
        for (int r = 0; r < 8; ++r) {
            // all-three >= 0  <=>  min3 >= 0  (v_min3_num_f32)
            float mn = fminf(fminf(du[r], dv[r]), dw[r]);
            int cand = (mn >= 0.0f) ? t : FID_NONE;
            fid[r] = min(fid[r], cand);
        }
    }

    // min over the 16 triangle columns (lanes within each 16-lane group)
#pragma unroll
    for (int r = 0; r < 8; ++r) {
        int v = fid[r];
        v = min(v, __shfl_xor(v, 1, 16));
        v = min(v, __shfl_xor(v, 2, 16));
        v = min(v, __shfl_xor(v, 4, 16));
        v = min(v, __shfl_xor(v, 8, 16));
        fid[r] = v;
    }
    if ((lane & 15) == 0) {
#pragma unroll
        for (int r = 0; r < 8; ++r) sfid[waveInBlk * 16 + 8 * g + r] = fid[r];
    }
    __syncthreads();

    // Interpolation: lane p (0..15) owns pixel pixBase+p; exact reference math.
    if (lane < 16) {
        const int p = pixBase + lane;
        if (p < total) {
            const int f = sfid[waveInBlk * 16 + lane];
            float o0 = 0.0f, o1 = 0.0f, o2 = 0.0f;
            if (f != FID_NONE) {
                const float qx = ((float)(p % res) + 0.5f) * invres;
                const float qy = ((float)(p / res) + 0.5f) * invres;
                const int ia = faces[3 * f + 0];
                const int ib = faces[3 * f + 1];
                const int ic = faces[3 * f + 2];
                const float ax = uv[2 * ia], ay = uv[2 * ia + 1];
                const float bx = uv[2 * ib], by = uv[2 * ib + 1];
                const float cx = uv[2 * ic], cy = uv[2 * ic + 1];
                const float v0x = bx - ax, v0y = by - ay;
                const float v1x = cx - ax, v1y = cy - ay;
                const float v2x = qx - ax, v2y = qy - ay;
                const float d00 = v0x * v0x + v0y * v0y;
                const float d01 = v0x * v1x + v0y * v1y;
                const float d11 = v1x * v1x + v1y * v1y;
                const float d20 = v2x * v0x + v2y * v0y;
                const float d21 = v2x * v1x + v2y * v1y;
                const float denom = d00 * d11 - d01 * d01;
                const bool bad = fabsf(denom) < BAKE_EPS;
                const float ds = bad ? 1.0f : denom;
                float bv = (d11 * d20 - d01 * d21) / ds;
                float bw = (d00 * d21 - d01 * d20) / ds;
                float bu = 1.0f - bv - bw;
                if (bad) { bu = bv = bw = -1.0f; }
                const float* aa = attr + 3 * (size_t)ia;
                const float* ab = attr + 3 * (size_t)ib;
                const float* ac = attr + 3 * (size_t)ic;
                o0 = bu * aa[0] + bv * ab[0] + bw * ac[0];
                o1 = bu * aa[1] + bv * ab[1] + bw * ac[1];
                o2 = bu * aa[2] + bv * ab[2] + bw * ac[2];
            }
            out[3 * (size_t)p + 0] = o0;
            out[3 * (size_t)p + 1] = o1;
            out[3 * (size_t)p + 2] = o2;
        }
    }
}

extern "C" void kernel_launch(void* const* d_in, const int* in_sizes, int n_in,
                              void* d_out, int out_size, void* d_ws, size_t ws_size,
                              hipStream_t stream) {
    const float* attr  = (const float*)d_in[0];
    const float* uv    = (const float*)d_in[1];
    const int*   faces = (const int*)d_in[2];
    float* out = (float*)d_out;
    float* coef = (float*)d_ws;

    const int nf = in_sizes[2] / 3;
    const int nTri = (nf / 64) * 64;   // reference truncates to full 64-chunks
    const int res = (int)(sqrt((double)out_size / 3.0) + 0.5);

    if (nTri > 0) {
        const int cb = (nTri + 255) / 256;
        bake_coef_kernel<<<cb, 256, 0, stream>>>(uv, faces, coef, nTri);
    }
    const int pixels = res * res;
    const int waves = (pixels + 15) / 16;
    const int blocks = (waves + 7) / 8;    // 8 waves (256 threads) per block
    bake_raster_kernel<<<blocks, 256, 0, stream>>>(attr, uv, faces, coef, out,
                                                   res, nTri);
}